// APHetNet_67276367724619
// MI455X (gfx1250) — compile-verified
//
#include <hip/hip_runtime.h>

typedef float v2f __attribute__((ext_vector_type(2)));
typedef float v4f __attribute__((ext_vector_type(4)));
typedef float v8f __attribute__((ext_vector_type(8)));

#define WAVES 4  // 128-thread blocks = 4 wave32s, each owns one 16-row tile

__device__ __forceinline__ float lrelu(float x) { return x >= 0.0f ? x : 0.1f * x; }

// ---------------------------------------------------------------------------
// Per-edge GEMM with V_WMMA_F32_16X16X4_F32.
// A row (per edge e) = [ xsrc[src_idx[e]][0:D0] | ea[e][0:D1] | xdst[dst_idx[e]][0:D2] ]
// All node/edge feature buffers are 64-float wide; D0/D1/D2 in {0,4,64}.
// WT is the transposed weight: WT[col*FI + k] = W[k*FO + col]  (v2f-loadable)
// MODE 0 (message): y = lrelu(A@W + b), atomic scatter-add into agg[dst_idx[e]*32+col] (FO=32)
// MODE 1 (edge-weight): y = lrelu(A@W + b) -> ea_out[e*64 + 4 + col] (FO=60), cols 0..3 untouched
// ---------------------------------------------------------------------------
template <int D0, int D1, int D2, int FO, int MODE>
__global__ void edge_gemm_kernel(const float* __restrict__ xsrc,
                                 const float* __restrict__ ea,
                                 const float* __restrict__ xdst,
                                 const int* __restrict__ src_idx,
                                 const int* __restrict__ dst_idx,
                                 const float* __restrict__ WT,
                                 const float* __restrict__ bias,
                                 float* __restrict__ out) {
  constexpr int FI  = D0 + D1 + D2;
  constexpr int LDA = FI + 4;                 // pad: rows stay 16B aligned, fewer bank conflicts
  constexpr int FOP = (FO + 15) & ~15;
  __shared__ float As[WAVES][16][LDA];

  const int lane = threadIdx.x & 31;
  const int wid  = threadIdx.x >> 5;
  const int e0   = (blockIdx.x * WAVES + wid) * 16;

  // ---- stage concatenated A rows into LDS with 16B vector ops ----
  {
    constexpr int V0 = D0 / 4;
    for (int i = lane; i < 16 * V0; i += 32) {
      int r = i / V0, kv = i % V0;
      *(v4f*)&As[wid][r][kv * 4] =
          *(const v4f*)&xsrc[src_idx[e0 + r] * 64 + kv * 4];
    }
  }
  {
    constexpr int V1 = D1 / 4;
    for (int i = lane; i < 16 * V1; i += 32) {
      int r = i / V1, kv = i % V1;
      *(v4f*)&As[wid][r][D0 + kv * 4] =
          *(const v4f*)&ea[(e0 + r) * 64 + kv * 4];
    }
  }
  if (D2 > 0) {
    constexpr int V2 = (D2 > 0) ? D2 / 4 : 1;
    for (int i = lane; i < 16 * V2; i += 32) {
      int r = i / V2, kv = i % V2;
      *(v4f*)&As[wid][r][D0 + D1 + kv * 4] =
          *(const v4f*)&xdst[dst_idx[e0 + r] * 64 + kv * 4];
    }
  }
  __syncthreads();

  // WMMA f32 16x16x4 fragment indexing (wave32):
  //   A: lane L holds M=L%16, K = kb + 2*(L/16) + {0,1}
  //   B: lane L holds N=L%16, K = kb + 2*(L/16) + {0,1}
  //   D: elem r, lane L -> M = r + 8*(L/16), N = L%16
  const int mrow  = lane & 15;
  const int koff  = (lane >> 4) << 1;
  const int rbase = (lane >> 4) << 3;

  for (int n0 = 0; n0 < FOP; n0 += 16) {
    const int col = n0 + (lane & 15);
    const float* wtc = WT + col * FI;         // transposed weight column (contiguous in K)
    v8f acc = {0.f, 0.f, 0.f, 0.f, 0.f, 0.f, 0.f, 0.f};
    for (int kb = 0; kb < FI; kb += 4) {
      v2f a = *(const v2f*)&As[wid][mrow][kb + koff];
      v2f b;
      if (FO == FOP || col < FO) {
        b = *(const v2f*)&wtc[kb + koff];     // single global_load_b64
      } else {
        b.x = 0.f; b.y = 0.f;
      }
      acc = __builtin_amdgcn_wmma_f32_16x16x4_f32(false, a, false, b,
                                                  (short)0, acc, false, false);
    }
    const float bv = (FO == FOP || col < FO) ? bias[col] : 0.0f;
#pragma unroll
    for (int r = 0; r < 8; ++r) {
      const int e = e0 + rbase + r;
      float v = lrelu(acc[r] + bv);
      if (MODE == 0) {
        atomicAdd(&out[dst_idx[e] * 32 + col], v);   // segment_sum
      } else {
        if (FO == FOP || col < FO) out[e * 64 + 4 + col] = v;
      }
    }
  }
}

// ---------------------------------------------------------------------------
// Node update: xout[:,4:64] = lrelu([xdst[:,0:DD] | agg] @ W + b) (+ xdst[:,4:64] if RES)
//              xout[:,0:4]  = xdst[:,0:4]
// ---------------------------------------------------------------------------
template <int DD, bool RES>
__global__ void node_upd_kernel(const float* __restrict__ xdst,
                                const float* __restrict__ agg,
                                const float* __restrict__ WT,
                                const float* __restrict__ bias,
                                float* __restrict__ xout) {
  constexpr int FI  = DD + 32;
  constexpr int LDA = FI + 4;
  __shared__ float As[WAVES][16][LDA];

  const int lane = threadIdx.x & 31;
  const int wid  = threadIdx.x >> 5;
  const int nd0  = (blockIdx.x * WAVES + wid) * 16;

  {
    constexpr int V0 = DD / 4;
    for (int i = lane; i < 16 * V0; i += 32) {
      int r = i / V0, kv = i % V0;
      *(v4f*)&As[wid][r][kv * 4] = *(const v4f*)&xdst[(nd0 + r) * 64 + kv * 4];
    }
  }
  for (int i = lane; i < 16 * 8; i += 32) {   // agg: 32 floats = 8 vec4 per row
    int r = i >> 3, kv = i & 7;
    *(v4f*)&As[wid][r][DD + kv * 4] = *(const v4f*)&agg[(nd0 + r) * 32 + kv * 4];
  }
  __syncthreads();

  const int mrow  = lane & 15;
  const int koff  = (lane >> 4) << 1;
  const int rbase = (lane >> 4) << 3;

  for (int n0 = 0; n0 < 64; n0 += 16) {
    const int col = n0 + (lane & 15);
    const float* wtc = WT + col * FI;
    v8f acc = {0.f, 0.f, 0.f, 0.f, 0.f, 0.f, 0.f, 0.f};
    for (int kb = 0; kb < FI; kb += 4) {
      v2f a = *(const v2f*)&As[wid][mrow][kb + koff];
      v2f b;
      if (col < 60) {
        b = *(const v2f*)&wtc[kb + koff];
      } else {
        b.x = 0.f; b.y = 0.f;
      }
      acc = __builtin_amdgcn_wmma_f32_16x16x4_f32(false, a, false, b,
                                                  (short)0, acc, false, false);
    }
    if (col < 60) {
      const float bv = bias[col];
#pragma unroll
      for (int r = 0; r < 8; ++r) {
        const int node = nd0 + rbase + r;
        float v = lrelu(acc[r] + bv);
        if (RES) v += xdst[node * 64 + 4 + col];
        xout[node * 64 + 4 + col] = v;
      }
    }
  }
  // copy first D=4 columns unchanged
  for (int i = lane; i < 16 * 4; i += 32) {
    int r = i >> 2, k = i & 3;
    xout[(nd0 + r) * 64 + k] = xdst[(nd0 + r) * 64 + k];
  }
}

// ---------------------------------------------------------------------------
// Small helpers
// ---------------------------------------------------------------------------
__global__ void transpose_w_kernel(const float* __restrict__ W, float* __restrict__ WT,
                                   int fi, int fo) {  // WT[c*fi + k] = W[k*fo + c]
  int i = blockIdx.x * 256 + threadIdx.x;
  if (i >= fi * fo) return;
  int k = i / fo, c = i % fo;
  WT[c * fi + k] = W[i];
}

__global__ void pack64_kernel(const float* __restrict__ src, float* __restrict__ dst,
                              int d, int total) {  // dst[row*64+k] = k<d ? src[row*d+k] : 0
  int i = blockIdx.x * 256 + threadIdx.x;
  if (i >= total) return;
  int row = i >> 6, k = i & 63;
  dst[i] = (k < d) ? src[row * d + k] : 0.0f;
}

__global__ void zero_kernel(float* __restrict__ p, int n) {
  int i = blockIdx.x * 256 + threadIdx.x;
  if (i < n) p[i] = 0.0f;
}

__global__ void copy_kernel(const float* __restrict__ s, float* __restrict__ d, int n) {
  int i = blockIdx.x * 256 + threadIdx.x;
  if (i < n) d[i] = s[i];
}

// Power head: h = lrelu(LN(ea@W1 + b1)); p = h@W2 + b2; out = [ea[:,0:4], p]
__global__ void power_kernel(const float* __restrict__ ead,
                             const float* __restrict__ W1, const float* __restrict__ b1,
                             const float* __restrict__ W2, const float* __restrict__ b2,
                             const float* __restrict__ g,  const float* __restrict__ bb,
                             float* __restrict__ out) {
  int e = blockIdx.x * 256 + threadIdx.x;
  const float* row = ead + (size_t)e * 64;
  float h0 = b1[0], h1 = b1[1], h2 = b1[2], h3 = b1[3];
#pragma unroll 8
  for (int k = 0; k < 64; ++k) {
    float x = row[k];
    h0 += x * W1[k * 4 + 0];
    h1 += x * W1[k * 4 + 1];
    h2 += x * W1[k * 4 + 2];
    h3 += x * W1[k * 4 + 3];
  }
  float mu = 0.25f * (h0 + h1 + h2 + h3);
  float d0 = h0 - mu, d1 = h1 - mu, d2 = h2 - mu, d3 = h3 - mu;
  float var = 0.25f * (d0 * d0 + d1 * d1 + d2 * d2 + d3 * d3);
  float rs = rsqrtf(var + 1e-5f);
  h0 = lrelu(d0 * rs * g[0] + bb[0]);
  h1 = lrelu(d1 * rs * g[1] + bb[1]);
  h2 = lrelu(d2 * rs * g[2] + bb[2]);
  h3 = lrelu(d3 * rs * g[3] + bb[3]);
  float p = b2[0] + h0 * W2[0] + h1 * W2[1] + h2 * W2[2] + h3 * W2[3];
  out[e * 5 + 0] = row[0];
  out[e * 5 + 1] = row[1];
  out[e * 5 + 2] = row[2];
  out[e * 5 + 3] = row[3];
  out[e * 5 + 4] = p;
}

// ---------------------------------------------------------------------------
extern "C" void kernel_launch(void* const* d_in, const int* in_sizes, int n_in,
                              void* d_out, int out_size, void* d_ws, size_t ws_size,
                              hipStream_t stream) {
  (void)n_in; (void)out_size; (void)ws_size;
  const float* x_ue     = (const float*)d_in[0];
  const float* x_ap     = (const float*)d_in[1];
  const float* ea_up_in = (const float*)d_in[2];
  const float* ea_dn_in = (const float*)d_in[3];
  const int* ei_up_src  = (const int*)d_in[4];
  const int* ei_up_dst  = (const int*)d_in[5];
  const int* ei_dn_src  = (const int*)d_in[6];
  const int* ei_dn_dst  = (const int*)d_in[7];
  const float* c1_msg_W = (const float*)d_in[8];   const float* c1_msg_b = (const float*)d_in[9];
  const float* c1_upd_W = (const float*)d_in[10];  const float* c1_upd_b = (const float*)d_in[11];
  const float* c1_ew_W  = (const float*)d_in[12];  const float* c1_ew_b  = (const float*)d_in[13];
  const float* c2_msg_W = (const float*)d_in[14];  const float* c2_msg_b = (const float*)d_in[15];
  const float* c2_upd_W = (const float*)d_in[16];  const float* c2_upd_b = (const float*)d_in[17];
  const float* c2_ew_W  = (const float*)d_in[18];  const float* c2_ew_b  = (const float*)d_in[19];
  const float* c3u_msg_W = (const float*)d_in[20]; const float* c3u_msg_b = (const float*)d_in[21];
  const float* c3u_upd_W = (const float*)d_in[22]; const float* c3u_upd_b = (const float*)d_in[23];
  const float* c3u_ew_W  = (const float*)d_in[24]; const float* c3u_ew_b  = (const float*)d_in[25];
  const float* c3d_msg_W = (const float*)d_in[26]; const float* c3d_msg_b = (const float*)d_in[27];
  const float* c3d_upd_W = (const float*)d_in[28]; const float* c3d_upd_b = (const float*)d_in[29];
  const float* c3d_ew_W  = (const float*)d_in[30]; const float* c3d_ew_b  = (const float*)d_in[31];
  const float* pe1_W = (const float*)d_in[32];     const float* pe1_b = (const float*)d_in[33];
  const float* pe2_W = (const float*)d_in[34];     const float* pe2_b = (const float*)d_in[35];
  const float* ln_g  = (const float*)d_in[36];     const float* ln_b  = (const float*)d_in[37];

  const int NUE = in_sizes[0] / 4;   // 512
  const int NAP = in_sizes[1] / 4;   // 1024
  const int E   = in_sizes[4];       // 524288 (divisible by 64 -> exact grids, EXEC always full)

  // workspace layout (floats)
  float* ws    = (float*)d_ws;
  float* ea_dn = ws;                              // [E,64] working ea_down
  float* xue0  = ea_dn + (size_t)E * 64;          // [NUE,64]
  float* xue1  = xue0 + (size_t)NUE * 64;         // [NUE,64]
  float* xap0  = xue1 + (size_t)NUE * 64;         // [NAP,64]
  float* xap1  = xap0 + (size_t)NAP * 64;         // [NAP,64]
  float* agg   = xap1 + (size_t)NAP * 64;         // [max(NAP,NUE),32]
  float* wtp   = agg + (size_t)NAP * 32;          // transposed weights

  const int TB = 256;
  // allocate + fill transposed weights
  float* wt[12];
  const float* wsrc[12] = {c1_msg_W, c1_upd_W, c1_ew_W, c2_msg_W, c2_upd_W, c2_ew_W,
                           c3u_msg_W, c3u_upd_W, c3u_ew_W, c3d_msg_W, c3d_upd_W, c3d_ew_W};
  const int wfi[12] = {8, 36, 12, 68, 36, 72, 128, 96, 192, 128, 96, 192};
  const int wfo[12] = {32, 60, 60, 32, 60, 60, 32, 60, 60, 32, 60, 60};
  for (int i = 0; i < 12; ++i) {
    wt[i] = wtp;
    wtp += (((size_t)wfi[i] * wfo[i]) + 3) & ~(size_t)3;
    int n = wfi[i] * wfo[i];
    transpose_w_kernel<<<(n + TB - 1) / TB, TB, 0, stream>>>(wsrc[i], wt[i], wfi[i], wfo[i]);
  }

  // output layout (floats): x_ue | x_ap | ea_up | ea_down
  float* out    = (float*)d_out;
  float* o_xue  = out;
  float* o_xap  = out + (size_t)NUE * 64;
  float* ea_up  = o_xap + (size_t)NAP * 64;       // ea_up working buffer lives in d_out
  float* o_ead  = ea_up + (size_t)E * 64;

  const dim3 eg((unsigned)(E / 64));              // 4 waves/block * 16 edges/wave
  const unsigned pb = (unsigned)(((size_t)E * 64) / TB);

  // init: widen raw features to 64-wide buffers
  pack64_kernel<<<(NUE * 64 + TB - 1) / TB, TB, 0, stream>>>(x_ue, xue0, 4, NUE * 64);
  pack64_kernel<<<(NAP * 64 + TB - 1) / TB, TB, 0, stream>>>(x_ap, xap0, 4, NAP * 64);
  pack64_kernel<<<pb, TB, 0, stream>>>(ea_up_in, ea_up, 4, E * 64);
  pack64_kernel<<<pb, TB, 0, stream>>>(ea_dn_in, ea_dn, 4, E * 64);

  // ---- conv1: UE --up--> AP ----
  zero_kernel<<<(NAP * 32 + TB - 1) / TB, TB, 0, stream>>>(agg, NAP * 32);
  edge_gemm_kernel<4, 4, 0, 32, 0><<<eg, 128, 0, stream>>>(
      xue0, ea_up, nullptr, ei_up_src, ei_up_dst, wt[0], c1_msg_b, agg);
  node_upd_kernel<4, false><<<NAP / 64, 128, 0, stream>>>(xap0, agg, wt[1], c1_upd_b, xap1);
  edge_gemm_kernel<4, 4, 4, 60, 1><<<eg, 128, 0, stream>>>(
      xue0, ea_up, xap0, ei_up_src, ei_up_dst, wt[2], c1_ew_b, ea_up);

  // ---- conv2: AP --down--> UE ----
  zero_kernel<<<(NUE * 32 + TB - 1) / TB, TB, 0, stream>>>(agg, NUE * 32);
  edge_gemm_kernel<64, 4, 0, 32, 0><<<eg, 128, 0, stream>>>(
      xap1, ea_dn, nullptr, ei_dn_src, ei_dn_dst, wt[3], c2_msg_b, agg);
  node_upd_kernel<4, false><<<NUE / 64, 128, 0, stream>>>(xue0, agg, wt[4], c2_upd_b, xue1);
  edge_gemm_kernel<64, 4, 4, 60, 1><<<eg, 128, 0, stream>>>(
      xap1, ea_dn, xue0, ei_dn_src, ei_dn_dst, wt[5], c2_ew_b, ea_dn);

  // ---- conv3 up: UE --up--> AP (residual) ----
  zero_kernel<<<(NAP * 32 + TB - 1) / TB, TB, 0, stream>>>(agg, NAP * 32);
  edge_gemm_kernel<64, 64, 0, 32, 0><<<eg, 128, 0, stream>>>(
      xue1, ea_up, nullptr, ei_up_src, ei_up_dst, wt[6], c3u_msg_b, agg);
  node_upd_kernel<64, true><<<NAP / 64, 128, 0, stream>>>(xap1, agg, wt[7], c3u_upd_b, xap0);
  edge_gemm_kernel<64, 64, 64, 60, 1><<<eg, 128, 0, stream>>>(
      xue1, ea_up, xap1, ei_up_src, ei_up_dst, wt[8], c3u_ew_b, ea_up);

  // ---- conv3 down: AP --down--> UE (residual) ----
  zero_kernel<<<(NUE * 32 + TB - 1) / TB, TB, 0, stream>>>(agg, NUE * 32);
  edge_gemm_kernel<64, 64, 0, 32, 0><<<eg, 128, 0, stream>>>(
      xap0, ea_dn, nullptr, ei_dn_src, ei_dn_dst, wt[9], c3d_msg_b, agg);
  node_upd_kernel<64, true><<<NUE / 64, 128, 0, stream>>>(xue1, agg, wt[10], c3d_upd_b, xue0);
  edge_gemm_kernel<64, 64, 64, 60, 1><<<eg, 128, 0, stream>>>(
      xap0, ea_dn, xue1, ei_dn_src, ei_dn_dst, wt[11], c3d_ew_b, ea_dn);

  // ---- power head + final outputs ----
  power_kernel<<<E / 256, 256, 0, stream>>>(ea_dn, pe1_W, pe1_b, pe2_W, pe2_b, ln_g, ln_b, o_ead);
  copy_kernel<<<(NUE * 64 + TB - 1) / TB, TB, 0, stream>>>(xue0, o_xue, NUE * 64);
  copy_kernel<<<(NAP * 64 + TB - 1) / TB, TB, 0, stream>>>(xap0, o_xap, NAP * 64);
}